// GAT_26731876450725
// MI455X (gfx1250) — compile-verified
//
#include <hip/hip_runtime.h>
#include <hip/hip_bf16.h>

// ---------------- problem constants (match reference) ----------------
#define N_NODES 50000
#define N_EDGES 800000
#define E_TOT   (N_EDGES + N_NODES)   // with self loops
#define F_IN    256
#define HEADS   8
#define HID     32
#define N_CLS   16
#define HC1     (HEADS * HID)   // 256
#define HC2     (HEADS * N_CLS) // 128
#define NEG_SLOPE 0.2f

typedef float v2f __attribute__((ext_vector_type(2)));
typedef float v8f __attribute__((ext_vector_type(8)));

// ---------------- helpers ----------------
__device__ __forceinline__ unsigned f2key(float f) {
  unsigned b = __float_as_uint(f);
  return (b & 0x80000000u) ? ~b : (b | 0x80000000u);
}
__device__ __forceinline__ float key2f(unsigned k) {
  return __uint_as_float((k & 0x80000000u) ? (k ^ 0x80000000u) : ~k);
}

// all buffers we zero have element counts divisible by 4
__global__ void zero4_kernel(float4* __restrict__ p, long long n4) {
  long long i = (long long)blockIdx.x * blockDim.x + threadIdx.x;
  if (i < n4) p[i] = make_float4(0.f, 0.f, 0.f, 0.f);
}

// ---------------- dense GEMM via V_WMMA_F32_16X16X4_F32 ----------------
// C[M,N] = A[M,K] @ B[K,N], f32.  One wave32 computes a 16x64 tile with four
// accumulators, reusing the A fragment 4x per K-step (2.25 loads / WMMA).
// A operand layout (16x4 f32): lanes 0-15 hold row M=lane, VGPR0=K0,VGPR1=K1;
// lanes 16-31 hold K2,K3.  B (4x16) mirrors with N=lane&15.
// C/D layout: VGPR r -> row r (lanes 0-15) / row r+8 (lanes 16-31), col=lane&15.
template <int N, int K>
__global__ __launch_bounds__(256) void gemm_f32_wmma(
    const float* __restrict__ A, const float* __restrict__ B,
    float* __restrict__ C, int M) {
  constexpr int TILES_N = N / 64;
  const int lane  = threadIdx.x & 31;
  const int wave  = blockIdx.x * (blockDim.x >> 5) + (threadIdx.x >> 5);
  const int tileM = wave / TILES_N;
  const int tileN = wave % TILES_N;
  if (tileM * 16 >= M) return;  // uniform per wave: EXEC stays all-ones

  const int r    = lane & 15;
  const int ksel = (lane >> 4) << 1;  // 0 or 2

  const float* __restrict__ arow  = A + (size_t)(tileM * 16 + r) * K + ksel;
  const float* __restrict__ bbase = B + (size_t)ksel * N + tileN * 64 + r;

  v8f acc0 = {}, acc1 = {}, acc2 = {}, acc3 = {};
  #pragma unroll 2
  for (int kk = 0; kk < K; kk += 4) {
    v2f a, b0, b1, b2, b3;
    a[0]  = arow[0];
    a[1]  = arow[1];
    b0[0] = bbase[0];       b0[1] = bbase[N];
    b1[0] = bbase[16];      b1[1] = bbase[N + 16];
    b2[0] = bbase[32];      b2[1] = bbase[N + 32];
    b3[0] = bbase[48];      b3[1] = bbase[N + 48];
    acc0 = __builtin_amdgcn_wmma_f32_16x16x4_f32(false, a, false, b0, (short)0, acc0, false, false);
    acc1 = __builtin_amdgcn_wmma_f32_16x16x4_f32(false, a, false, b1, (short)0, acc1, false, false);
    acc2 = __builtin_amdgcn_wmma_f32_16x16x4_f32(false, a, false, b2, (short)0, acc2, false, false);
    acc3 = __builtin_amdgcn_wmma_f32_16x16x4_f32(false, a, false, b3, (short)0, acc3, false, false);
    arow  += 4;
    bbase += (size_t)4 * N;
  }

  float* __restrict__ crow =
      C + (size_t)(tileM * 16 + (lane >> 4) * 8) * N + tileN * 64 + (lane & 15);
  #pragma unroll
  for (int i = 0; i < 8; ++i) {
    crow[(size_t)i * N +  0] = acc0[i];
    crow[(size_t)i * N + 16] = acc1[i];
    crow[(size_t)i * N + 32] = acc2[i];
    crow[(size_t)i * N + 48] = acc3[i];
  }
}

// ---------------- per-node attention logits ----------------
// as[n,h] = dot(h[n,h,:], a_src[h,:]); ad likewise. One thread per (n,head).
template <int C>
__global__ void alpha_kernel(const float* __restrict__ h,
                             const float* __restrict__ a_src,
                             const float* __restrict__ a_dst,
                             float* __restrict__ as, float* __restrict__ ad) {
  int idx = blockIdx.x * blockDim.x + threadIdx.x;
  if (idx >= N_NODES * HEADS) return;
  int n = idx >> 3, hd = idx & 7;
  const float* hp = h + (size_t)n * HEADS * C + (size_t)hd * C;
  const float* av = a_src + hd * C;
  const float* bv = a_dst + hd * C;
  float s = 0.f, d = 0.f;
  #pragma unroll 4
  for (int c = 0; c < C; ++c) {
    float v = hp[c];
    s += v * av[c];
    d += v * bv[c];
  }
  as[idx] = s;
  ad[idx] = d;
}

__device__ __forceinline__ void edge_endpoints(const int* __restrict__ ei,
                                               int e, int& src, int& dst) {
  if (e < N_EDGES) {
    src = ei[e];
    dst = ei[N_EDGES + e];
  } else {
    src = dst = e - N_EDGES;  // self loop
  }
}

// ---------------- pass A: leaky-relu logits + segment max ----------------
__global__ void edge_max_kernel(const int* __restrict__ ei,
                                const float* __restrict__ as,
                                const float* __restrict__ ad,
                                unsigned* __restrict__ mkeys) {
  long long idx = (long long)blockIdx.x * blockDim.x + threadIdx.x;
  if (idx >= (long long)E_TOT * HEADS) return;
  int e = (int)(idx >> 3), hd = (int)(idx & 7);
  int src, dst;
  edge_endpoints(ei, e, src, dst);
  float v = as[src * HEADS + hd] + ad[dst * HEADS + hd];
  v = v > 0.f ? v : NEG_SLOPE * v;
  atomicMax(&mkeys[dst * HEADS + hd], f2key(v));
}

// ---------------- pass B: exp + segment sum ----------------
__global__ void edge_exp_kernel(const int* __restrict__ ei,
                                const float* __restrict__ as,
                                const float* __restrict__ ad,
                                const unsigned* __restrict__ mkeys,
                                float* __restrict__ ex,
                                float* __restrict__ den) {
  long long idx = (long long)blockIdx.x * blockDim.x + threadIdx.x;
  if (idx >= (long long)E_TOT * HEADS) return;
  int e = (int)(idx >> 3), hd = (int)(idx & 7);
  int src, dst;
  edge_endpoints(ei, e, src, dst);
  float v = as[src * HEADS + hd] + ad[dst * HEADS + hd];
  v = v > 0.f ? v : NEG_SLOPE * v;
  float m  = key2f(mkeys[dst * HEADS + hd]);
  float xv = __expf(v - m);
  ex[idx]  = xv;
  atomicAdd(&den[dst * HEADS + hd], xv);
}

// ---------------- pass C: weighted gather/scatter aggregation ----------------
// One thread per (edge, 4 channels): b128 gather of h[src], 4 f32 atomics to
// agg[dst]. blockDim=256 covers whole edges -> fully coalesced segments.
template <int HCv, int Cv>
__global__ void edge_agg_kernel(const int* __restrict__ ei,
                                const float* __restrict__ ex,
                                const float* __restrict__ den,
                                const float* __restrict__ hmat,
                                float* __restrict__ agg) {
  constexpr int HC4 = HCv / 4;
  long long idx = (long long)blockIdx.x * blockDim.x + threadIdx.x;
  if (idx >= (long long)E_TOT * HC4) return;
  int e = (int)(idx / HC4);
  int j = (int)(idx % HC4) * 4;   // Cv is 16 or 32 -> all 4 lanes same head
  int hd = j / Cv;
  int src, dst;
  edge_endpoints(ei, e, src, dst);
  float w = ex[(size_t)e * HEADS + hd] /
            (den[(size_t)dst * HEADS + hd] + 1e-16f);
  const float4 hv = *(const float4*)(hmat + (size_t)src * HCv + j);
  float* ap = agg + (size_t)dst * HCv + j;
  atomicAdd(ap + 0, w * hv.x);
  atomicAdd(ap + 1, w * hv.y);
  atomicAdd(ap + 2, w * hv.z);
  atomicAdd(ap + 3, w * hv.w);
}

// ---------------- epilogues ----------------
template <int HCv>
__global__ void bias_relu_kernel(float* __restrict__ p,
                                 const float* __restrict__ bias, long long n) {
  long long i = (long long)blockIdx.x * blockDim.x + threadIdx.x;
  if (i >= n) return;
  float v = p[i] + bias[(int)(i % HCv)];
  p[i] = v > 0.f ? v : 0.f;
}

template <int HCv>
__global__ void bias_add_kernel(float* __restrict__ p,
                                const float* __restrict__ bias, long long n) {
  long long i = (long long)blockIdx.x * blockDim.x + threadIdx.x;
  if (i >= n) return;
  p[i] = p[i] + bias[(int)(i % HCv)];
}

// ---------------- host-side orchestration ----------------
static inline int blocks_for(long long n, int bs) {
  return (int)((n + bs - 1) / bs);
}

extern "C" void kernel_launch(void* const* d_in, const int* in_sizes, int n_in,
                              void* d_out, int out_size, void* d_ws, size_t ws_size,
                              hipStream_t stream) {
  const float* x      = (const float*)d_in[0];
  const int*   ei     = (const int*)  d_in[1];
  const float* W1     = (const float*)d_in[2];
  const float* a_src1 = (const float*)d_in[3];
  const float* a_dst1 = (const float*)d_in[4];
  const float* b1     = (const float*)d_in[5];
  const float* W2     = (const float*)d_in[6];
  const float* a_src2 = (const float*)d_in[7];
  const float* a_dst2 = (const float*)d_in[8];
  const float* b2     = (const float*)d_in[9];
  float* out = (float*)d_out;

  // ---- workspace layout (floats) ----
  char* ws = (char*)d_ws;
  size_t off = 0;
  auto carve = [&](size_t bytes) {
    void* p = ws + off;
    off += (bytes + 255) & ~(size_t)255;
    return p;
  };
  float*    hbuf   = (float*)   carve((size_t)N_NODES * HC1 * 4); // h1, reused for h2
  float*    aggbuf = (float*)   carve((size_t)N_NODES * HC1 * 4); // layer-1 agg -> relu'd feats
  float*    as     = (float*)   carve((size_t)N_NODES * HEADS * 4);
  float*    ad     = (float*)   carve((size_t)N_NODES * HEADS * 4);
  unsigned* mkeys  = (unsigned*)carve((size_t)N_NODES * HEADS * 4);
  float*    den    = (float*)   carve((size_t)N_NODES * HEADS * 4);
  float*    ex     = (float*)   carve((size_t)E_TOT   * HEADS * 4);

  const int BS = 256;
  const long long nNH    = (long long)N_NODES * HEADS;
  const long long nEH    = (long long)E_TOT * HEADS;
  const long long nElem1 = (long long)N_NODES * HC1;
  const long long nElem2 = (long long)N_NODES * HC2;

  // =================== layer 1 ===================
  // h1 = x @ W1   [50000,256]x[256,256]
  {
    long long waves = (N_NODES / 16) * (HC1 / 64);
    gemm_f32_wmma<HC1, F_IN><<<blocks_for(waves * 32, BS), BS, 0, stream>>>(
        x, W1, hbuf, N_NODES);
  }
  alpha_kernel<HID><<<blocks_for(nNH, BS), BS, 0, stream>>>(hbuf, a_src1, a_dst1, as, ad);
  zero4_kernel<<<blocks_for(nNH / 4, BS), BS, 0, stream>>>((float4*)mkeys, nNH / 4);
  zero4_kernel<<<blocks_for(nNH / 4, BS), BS, 0, stream>>>((float4*)den, nNH / 4);
  edge_max_kernel<<<blocks_for(nEH, BS), BS, 0, stream>>>(ei, as, ad, mkeys);
  edge_exp_kernel<<<blocks_for(nEH, BS), BS, 0, stream>>>(ei, as, ad, mkeys, ex, den);
  zero4_kernel<<<blocks_for(nElem1 / 4, BS), BS, 0, stream>>>((float4*)aggbuf, nElem1 / 4);
  edge_agg_kernel<HC1, HID><<<blocks_for((long long)E_TOT * (HC1 / 4), BS), BS, 0, stream>>>(
      ei, ex, den, hbuf, aggbuf);
  bias_relu_kernel<HC1><<<blocks_for(nElem1, BS), BS, 0, stream>>>(aggbuf, b1, nElem1);

  // =================== layer 2 ===================
  // h2 = relu(agg1 + b1) @ W2   [50000,256]x[256,128]  (h2 overwrites hbuf)
  {
    long long waves = (N_NODES / 16) * (HC2 / 64);
    gemm_f32_wmma<HC2, HC1><<<blocks_for(waves * 32, BS), BS, 0, stream>>>(
        aggbuf, W2, hbuf, N_NODES);
  }
  alpha_kernel<N_CLS><<<blocks_for(nNH, BS), BS, 0, stream>>>(hbuf, a_src2, a_dst2, as, ad);
  zero4_kernel<<<blocks_for(nNH / 4, BS), BS, 0, stream>>>((float4*)mkeys, nNH / 4);
  zero4_kernel<<<blocks_for(nNH / 4, BS), BS, 0, stream>>>((float4*)den, nNH / 4);
  edge_max_kernel<<<blocks_for(nEH, BS), BS, 0, stream>>>(ei, as, ad, mkeys);
  edge_exp_kernel<<<blocks_for(nEH, BS), BS, 0, stream>>>(ei, as, ad, mkeys, ex, den);
  zero4_kernel<<<blocks_for(nElem2 / 4, BS), BS, 0, stream>>>((float4*)out, nElem2 / 4);
  edge_agg_kernel<HC2, N_CLS><<<blocks_for((long long)E_TOT * (HC2 / 4), BS), BS, 0, stream>>>(
      ei, ex, den, hbuf, out);
  bias_add_kernel<HC2><<<blocks_for(nElem2, BS), BS, 0, stream>>>(out, b2, nElem2);
}